// CrossEntropyLoss3_1211180778077
// MI455X (gfx1250) — compile-verified
//
#include <hip/hip_runtime.h>

// CrossEntropyLoss3 for MI455X (gfx1250, wave32).
//
// Roofline: 512 MB streamed once (predict + one-hot target) / 23.3 TB/s
// ~= 22 us -> strictly bandwidth-bound. Hot loop: 16 lanes cooperate on one
// 64-float row (float4/lane, nontemporal b128 loads), row max / sum-exp via
// 4-step xor-shuffles within each 16-lane half. For a one-hot target the
// contributing element j IS argmax(target row), so alpha_j = pm[j][j] — no
// cross-lane label reduction needed. Deterministic two-pass reduction (no
// float atomics -> bit-stable across graph replays). Finalize's last 64->1
// sum runs through the CDNA5 matrix pipe (v_wmma_f32_16x16x4_f32, B = ones).

typedef float f4  __attribute__((ext_vector_type(4)));
typedef float v2f __attribute__((ext_vector_type(2)));
typedef float v8f __attribute__((ext_vector_type(8)));

#define WCLS 64
#define NBLK 4096

__global__ void __launch_bounds__(256) ce3_rows(const float* __restrict__ pred,
                                                const float* __restrict__ targ,
                                                const float* __restrict__ pm,
                                                float* __restrict__ partial,
                                                int nrows) {
    const int lane = threadIdx.x & 31;
    const int sub  = threadIdx.x & 15;   // lane within 16-lane row group
    const long long half_id = ((long long)blockIdx.x * blockDim.x + threadIdx.x) >> 4;
    const long long nhalves = ((long long)gridDim.x * blockDim.x) >> 4;
    const size_t stride_elts = (size_t)nhalves * WCLS;

    float acc = 0.0f;

    for (long long row = half_id; row < nrows; row += nhalves) {
        const size_t base = (size_t)row * WCLS + (size_t)sub * 4;

        // Streaming, read-once data: non-temporal 128-bit loads (TH=NT keeps
        // the 512 MB stream from rinsing through the 192 MB L2).
        f4 v = __builtin_nontemporal_load((const f4*)(pred + base));
        f4 t = __builtin_nontemporal_load((const f4*)(targ + base));

        // Prefetch next iteration's lines (speculative; global_prefetch_b8).
        __builtin_prefetch(pred + base + stride_elts, 0, 1);
        __builtin_prefetch(targ + base + stride_elts, 0, 1);

        // ---- row max (16-lane butterfly; xor masks < 16 stay in the half) ----
        float m = fmaxf(fmaxf(v.x, v.y), fmaxf(v.z, v.w));
        #pragma unroll
        for (int mk = 8; mk >= 1; mk >>= 1)
            m = fmaxf(m, __shfl_xor(m, mk, 32));

        // ---- sum of exp(v - m) ----
        float e0 = __expf(v.x - m), e1 = __expf(v.y - m);
        float e2 = __expf(v.z - m), e3 = __expf(v.w - m);
        float s = (e0 + e1) + (e2 + e3);
        #pragma unroll
        for (int mk = 8; mk >= 1; mk >>= 1)
            s += __shfl_xor(s, mk, 32);

        // ---- loss: only the lane holding the one-hot's nonzero contributes.
        // For one-hot targets, a nonzero element j is argmax(target row), so
        // alpha_j = penalty_matrix[j][j] (== 1.0 by construction; loaded
        // anyway to stay faithful to the input matrix).
        if ((t.x != 0.0f) || (t.y != 0.0f) || (t.z != 0.0f) || (t.w != 0.0f)) {
            const float logs  = __logf(s);
            const float inv_s = 1.0f / s;
            const int   j0    = sub * 4;
            if (t.x != 0.0f) {
                float lp = (v.x - m) - logs, p = e0 * inv_s;
                acc += -lp * t.x * (__expf(1.0f - p * pm[(size_t)(j0 + 0) * (WCLS + 1)]) - 1.0f);
            }
            if (t.y != 0.0f) {
                float lp = (v.y - m) - logs, p = e1 * inv_s;
                acc += -lp * t.y * (__expf(1.0f - p * pm[(size_t)(j0 + 1) * (WCLS + 1)]) - 1.0f);
            }
            if (t.z != 0.0f) {
                float lp = (v.z - m) - logs, p = e2 * inv_s;
                acc += -lp * t.z * (__expf(1.0f - p * pm[(size_t)(j0 + 2) * (WCLS + 1)]) - 1.0f);
            }
            if (t.w != 0.0f) {
                float lp = (v.w - m) - logs, p = e3 * inv_s;
                acc += -lp * t.w * (__expf(1.0f - p * pm[(size_t)(j0 + 3) * (WCLS + 1)]) - 1.0f);
            }
        }
    }

    // ---- deterministic block reduction ----
    #pragma unroll
    for (int mk = 16; mk >= 1; mk >>= 1)
        acc += __shfl_xor(acc, mk, 32);

    __shared__ float wsum[8];
    if (lane == 0) wsum[threadIdx.x >> 5] = acc;
    __syncthreads();
    if (threadIdx.x == 0) {
        float b = 0.0f;
        #pragma unroll
        for (int i = 0; i < 8; ++i) b += wsum[i];
        partial[blockIdx.x] = b;
    }
}

// One-shot finalize: deterministic reduction of block partials; the final
// 64-value sum goes through the CDNA5 matrix pipe (v_wmma_f32_16x16x4_f32).
__global__ void __launch_bounds__(256) ce3_finalize(const float* __restrict__ partial,
                                                    int nparts,
                                                    float* __restrict__ out,
                                                    float inv_n) {
    __shared__ float red[256];
    float s = 0.0f;
    for (int i = threadIdx.x; i < nparts; i += 256) s += partial[i];
    red[threadIdx.x] = s;
    __syncthreads();
    if (threadIdx.x < 64) {
        red[threadIdx.x] = (red[threadIdx.x] + red[threadIdx.x + 64]) +
                           (red[threadIdx.x + 128] + red[threadIdx.x + 192]);
    }
    __syncthreads();
    if (threadIdx.x < 32) {                 // wave 0, EXEC all-1s (WMMA requirement)
        const int lane = threadIdx.x;
        const int m2   = lane & 15;         // A-matrix row M
        const int khi  = lane >> 4;         // K half: lanes 0-15 -> K0,K1 ; 16-31 -> K2,K3
        // A (16x4 f32): A[m][k] = red[m*4 + k], per ISA 32-bit A layout.
        v2f a;
        a.x = red[m2 * 4 + khi * 2 + 0];
        a.y = red[m2 * 4 + khi * 2 + 1];
        v2f b; b.x = 1.0f; b.y = 1.0f;      // B = all-ones (layout-independent)
        v8f c = {};
        v8f d = __builtin_amdgcn_wmma_f32_16x16x4_f32(
            false, a, false, b, (short)0, c, false, false);
        // D[m][n] = rowsum_m for every n. VGPR r: lanes 0-15 hold M=r, lanes
        // 16-31 hold M=r+8 -> per-lane sum covers M 0..7 (lo) or 8..15 (hi).
        float p8 = 0.0f;
        #pragma unroll
        for (int r = 0; r < 8; ++r) p8 += d[r];
        float tot = p8 + __shfl_xor(p8, 16, 32);   // all 16 row sums
        if (lane == 0) out[0] = tot * inv_n;
    }
}

extern "C" void kernel_launch(void* const* d_in, const int* in_sizes, int n_in,
                              void* d_out, int out_size, void* d_ws, size_t ws_size,
                              hipStream_t stream) {
    const float* pred = (const float*)d_in[0];
    const float* targ = (const float*)d_in[1];
    const float* pm   = (const float*)d_in[2];
    float* out        = (float*)d_out;
    float* partial    = (float*)d_ws;

    const int nrows = in_sizes[0] / WCLS;

    int nblocks = NBLK;
    size_t cap = ws_size / sizeof(float);
    if ((size_t)nblocks > cap) nblocks = (int)cap;
    if (nblocks < 1) nblocks = 1;

    ce3_rows<<<nblocks, 256, 0, stream>>>(pred, targ, pm, partial, nrows);
    ce3_finalize<<<1, 256, 0, stream>>>(partial, nblocks, out, 1.0f / (float)nrows);
}